// SparseCoder_20641612824956
// MI455X (gfx1250) — compile-verified
//
#include <hip/hip_runtime.h>
#include <hip/hip_bf16.h>

// ---------------------------------------------------------------------------
// ISTA sparse coding on gfx1250 (MI455X).
//   fp32 WMMA (v_wmma_f32_16x16x4_f32) + async global->LDS staging
//   (global_load_async_to_lds_b128 / s_wait_asynccnt) with double buffering.
//   x: [256,512]  D: [1024,512]  h(out): [256,1024]
// ---------------------------------------------------------------------------

#define BATCH 256
#define INPUT 512
#define DICT  1024
#define N_ITERS 1000
#define LMBD 0.1f
#define TOL 1e-4f
#define POWER_ITERS 128

#define KC 32            // K-chunk staged per double-buffer step
#define LDS_STRIDE 36    // 32 + 4 pad: 144B rows -> b128-aligned, conflict-free

typedef __attribute__((ext_vector_type(2))) float v2f;
typedef __attribute__((ext_vector_type(8))) float v8f;

// ---- workspace layout (in floats) ----
#define FOFF_SCAL  8
#define FOFF_ROWSQ 64
#define FOFF_DT    1024
#define FOFF_RESID (1024 + INPUT * DICT)

// ---------------------------------------------------------------------------
// Async copy of a 64-row x 32-float tile (global -> LDS), 256 threads.
// 512 b128 transfers -> 2 per thread. LDS rows padded to LDS_STRIDE floats.
// ---------------------------------------------------------------------------
__device__ __forceinline__ void async_tile_64x32(const float* __restrict__ src,
                                                 int ld, unsigned ldsBase, int tid) {
#pragma unroll
  for (int s = 0; s < 2; ++s) {
    const int e = tid + 256 * s;
    const int row = e >> 3;        // 0..63
    const int quad = e & 7;        // 0..7 (4 floats each)
    const float* g = src + row * ld + quad * 4;
    const unsigned l = ldsBase + (unsigned)((row * LDS_STRIDE + quad * 4) * 4);
    asm volatile("global_load_async_to_lds_b128 %0, %1, off"
                 :: "v"(l), "v"(g) : "memory");
  }
}

// ---------------------------------------------------------------------------
// One-time: Dt = D^T  (LDS-tiled, coalesced both directions)
// ---------------------------------------------------------------------------
__global__ void transpose_kernel(const float* __restrict__ D, float* __restrict__ Dt) {
  __shared__ float tile[16][17];
  const int j0 = blockIdx.x * 16;
  const int i0 = blockIdx.y * 16;
  const int tx = threadIdx.x, ty = threadIdx.y;
  tile[ty][tx] = D[(i0 + ty) * INPUT + (j0 + tx)];
  __syncthreads();
  Dt[(j0 + ty) * DICT + (i0 + tx)] = tile[tx][ty];
}

// ---------------------------------------------------------------------------
// One-time: power iteration on G = D D^T -> lr = 0.99 / lambda_max.
// ---------------------------------------------------------------------------
__global__ void power_kernel(const float* __restrict__ D, float* __restrict__ scal) {
  __shared__ float v[DICT];
  __shared__ float w[INPUT];
  __shared__ float red[32];
  const int t = threadIdx.x;  // 0..1023
  v[t] = 1.0f;
  __syncthreads();
  float lambda = 1.0f;
  for (int it = 0; it < POWER_ITERS; ++it) {
    if (t < INPUT) {
      float s = 0.0f;
      for (int i = 0; i < DICT; ++i) s += D[i * INPUT + t] * v[i];
      w[t] = s;
    }
    __syncthreads();
    float s = 0.0f;
    const float* drow = D + t * INPUT;
    for (int j = 0; j < INPUT; ++j) s += drow[j] * w[j];
    const float u = s;
    float sq = u * u;
    for (int off = 16; off > 0; off >>= 1) sq += __shfl_xor(sq, off, 32);
    if ((t & 31) == 0) red[t >> 5] = sq;
    __syncthreads();
    if (t < 32) {
      float s2 = red[t];
      for (int off = 16; off > 0; off >>= 1) s2 += __shfl_xor(s2, off, 32);
      if (t == 0) red[0] = s2;
    }
    __syncthreads();
    const float nrm = sqrtf(red[0]);
    lambda = nrm;
    v[t] = u / nrm;
    __syncthreads();
  }
  if (t == 0) {
    const float lr = 0.99f / lambda;
    scal[0] = lr;
    scal[1] = lr * LMBD;
  }
}

// ---------------------------------------------------------------------------
// resid = x - h @ D   (M=256, N=512, K=1024)
// Block: 256 thr (8 waves), 64x64 macro-tile, waves 4(M)x2(N), wave tile 16x32.
// A chunk = rows of h; Bt chunk = rows of Dt (Bt[n][k] = D[k][n]).
// ---------------------------------------------------------------------------
__global__ void resid_kernel(const float* __restrict__ x, const float* __restrict__ Dt,
                             const float* __restrict__ h, float* __restrict__ resid,
                             const int* __restrict__ flags) {
  if (flags[0]) return;  // uniform branch: EXEC all-1s for WMMA
  __shared__ __align__(16) float Alds[2][64][LDS_STRIDE];
  __shared__ __align__(16) float Blds[2][64][LDS_STRIDE];
  const int tid = threadIdx.x;
  const int lane = tid & 31, w = tid >> 5;
  const int half = lane >> 4, r = lane & 15;
  const int wm = w >> 1, wn = w & 1;
  const int gm0 = blockIdx.y * 64, gn0 = blockIdx.x * 64;
  const float* Abase = h + (size_t)gm0 * DICT;   // lda = 1024
  const float* Bbase = Dt + (size_t)gn0 * DICT;  // Bt rows, ld = 1024

  v8f acc0 = {}, acc1 = {};
  async_tile_64x32(Abase, DICT, (unsigned)(size_t)&Alds[0][0][0], tid);
  async_tile_64x32(Bbase, DICT, (unsigned)(size_t)&Blds[0][0][0], tid);

  const int NC = DICT / KC;  // 32 chunks
  for (int c = 0; c < NC; ++c) {
    const int buf = c & 1;
    if (c + 1 < NC) {
      const int kc = (c + 1) * KC;
      const int nb = buf ^ 1;
      async_tile_64x32(Abase + kc, DICT, (unsigned)(size_t)&Alds[nb][0][0], tid);
      async_tile_64x32(Bbase + kc, DICT, (unsigned)(size_t)&Blds[nb][0][0], tid);
      asm volatile("s_wait_asynccnt 0x4" ::: "memory");  // chunk c complete
    } else {
      asm volatile("s_wait_asynccnt 0x0" ::: "memory");
    }
    __syncthreads();
#pragma unroll
    for (int k0 = 0; k0 < KC; k0 += 4) {
      v2f a  = *(const v2f*)&Alds[buf][wm * 16 + r][k0 + 2 * half];
      v2f b0 = *(const v2f*)&Blds[buf][wn * 32 + r][k0 + 2 * half];
      v2f b1 = *(const v2f*)&Blds[buf][wn * 32 + 16 + r][k0 + 2 * half];
      acc0 = __builtin_amdgcn_wmma_f32_16x16x4_f32(false, a, false, b0,
                                                   (short)0, acc0, false, false);
      acc1 = __builtin_amdgcn_wmma_f32_16x16x4_f32(false, a, false, b1,
                                                   (short)0, acc1, false, false);
    }
    __syncthreads();
  }

#pragma unroll
  for (int jj = 0; jj < 2; ++jj) {
    const v8f a = jj ? acc1 : acc0;
    const int col = gn0 + wn * 32 + jj * 16 + r;
#pragma unroll
    for (int v = 0; v < 8; ++v) {
      const int row = gm0 + wm * 16 + v + 8 * half;
      resid[row * INPUT + col] = x[row * INPUT + col] - a[v];
    }
  }
}

// ---------------------------------------------------------------------------
// g = resid @ D^T; h = soft_threshold(h + lr*g); accumulate per-row ||dh||^2.
// (M=256, N=1024, K=512). Bt[n][k] = D[n][k] -> rows of D directly.
// ---------------------------------------------------------------------------
__global__ void update_kernel(const float* __restrict__ resid, const float* __restrict__ D,
                              float* __restrict__ h, const float* __restrict__ scal,
                              float* __restrict__ rowsq, const int* __restrict__ flags) {
  if (flags[0]) return;
  __shared__ __align__(16) float Alds[2][64][LDS_STRIDE];
  __shared__ __align__(16) float Blds[2][64][LDS_STRIDE];
  const int tid = threadIdx.x;
  const int lane = tid & 31, w = tid >> 5;
  const int half = lane >> 4, r = lane & 15;
  const int wm = w >> 1, wn = w & 1;
  const int gm0 = blockIdx.y * 64, gn0 = blockIdx.x * 64;
  const float* Abase = resid + (size_t)gm0 * INPUT;  // lda = 512
  const float* Bbase = D + (size_t)gn0 * INPUT;      // Bt rows = D rows, ld = 512

  v8f acc0 = {}, acc1 = {};
  async_tile_64x32(Abase, INPUT, (unsigned)(size_t)&Alds[0][0][0], tid);
  async_tile_64x32(Bbase, INPUT, (unsigned)(size_t)&Blds[0][0][0], tid);

  const int NC = INPUT / KC;  // 16 chunks
  for (int c = 0; c < NC; ++c) {
    const int buf = c & 1;
    if (c + 1 < NC) {
      const int kc = (c + 1) * KC;
      const int nb = buf ^ 1;
      async_tile_64x32(Abase + kc, INPUT, (unsigned)(size_t)&Alds[nb][0][0], tid);
      async_tile_64x32(Bbase + kc, INPUT, (unsigned)(size_t)&Blds[nb][0][0], tid);
      asm volatile("s_wait_asynccnt 0x4" ::: "memory");
    } else {
      asm volatile("s_wait_asynccnt 0x0" ::: "memory");
    }
    __syncthreads();
#pragma unroll
    for (int k0 = 0; k0 < KC; k0 += 4) {
      v2f a  = *(const v2f*)&Alds[buf][wm * 16 + r][k0 + 2 * half];
      v2f b0 = *(const v2f*)&Blds[buf][wn * 32 + r][k0 + 2 * half];
      v2f b1 = *(const v2f*)&Blds[buf][wn * 32 + 16 + r][k0 + 2 * half];
      acc0 = __builtin_amdgcn_wmma_f32_16x16x4_f32(false, a, false, b0,
                                                   (short)0, acc0, false, false);
      acc1 = __builtin_amdgcn_wmma_f32_16x16x4_f32(false, a, false, b1,
                                                   (short)0, acc1, false, false);
    }
    __syncthreads();
  }

  const float lr = scal[0];
  const float thresh = scal[1];
  float dsq[8];
#pragma unroll
  for (int v = 0; v < 8; ++v) dsq[v] = 0.0f;

#pragma unroll
  for (int jj = 0; jj < 2; ++jj) {
    const v8f a = jj ? acc1 : acc0;
    const int col = gn0 + wn * 32 + jj * 16 + r;
#pragma unroll
    for (int v = 0; v < 8; ++v) {
      const int row = gm0 + wm * 16 + v + 8 * half;
      const int idx = row * DICT + col;
      const float hOld = h[idx];
      const float aVal = hOld + lr * a[v];
      const float mag = fabsf(aVal) - thresh;
      const float hNew = (mag > 0.0f) ? copysignf(mag, aVal) : 0.0f;
      h[idx] = hNew;
      const float d = hNew - hOld;
      dsq[v] += d * d;
    }
  }
  // sum across the 16 columns of each half (xor masks 1..8 stay in-half)
#pragma unroll
  for (int v = 0; v < 8; ++v) {
    float s = dsq[v];
    for (int off = 1; off < 16; off <<= 1) s += __shfl_xor(s, off, 32);
    dsq[v] = s;
  }
  if (r == 0) {
#pragma unroll
    for (int v = 0; v < 8; ++v)
      atomicAdd(&rowsq[gm0 + wm * 16 + v + 8 * half], dsq[v]);
  }
}

// ---------------------------------------------------------------------------
// delta = max_row sqrt(rowsq); done = (delta < TOL). Zeroes rowsq.
// ---------------------------------------------------------------------------
__global__ void check_kernel(float* __restrict__ rowsq, int* __restrict__ flags) {
  if (flags[0]) return;
  __shared__ float smax[256];
  const int t = threadIdx.x;
  const float v = rowsq[t];
  rowsq[t] = 0.0f;
  smax[t] = v;
  __syncthreads();
  for (int s = 128; s > 0; s >>= 1) {
    if (t < s) smax[t] = fmaxf(smax[t], smax[t + s]);
    __syncthreads();
  }
  if (t == 0 && smax[0] < TOL * TOL) flags[0] = 1;
}

// ---------------------------------------------------------------------------
extern "C" void kernel_launch(void* const* d_in, const int* in_sizes, int n_in,
                              void* d_out, int out_size, void* d_ws, size_t ws_size,
                              hipStream_t stream) {
  (void)in_sizes; (void)n_in; (void)out_size; (void)ws_size;
  const float* x = (const float*)d_in[0];   // [256,512]
  const float* D = (const float*)d_in[1];   // [1024,512]
  float* h = (float*)d_out;                 // [256,1024]

  float* wsf   = (float*)d_ws;
  int*   flags = (int*)d_ws;
  float* scal  = wsf + FOFF_SCAL;
  float* rowsq = wsf + FOFF_ROWSQ;
  float* Dt    = wsf + FOFF_DT;
  float* resid = wsf + FOFF_RESID;

  hipMemsetAsync(d_out, 0, (size_t)BATCH * DICT * sizeof(float), stream);
  hipMemsetAsync(d_ws, 0, 4096, stream);

  transpose_kernel<<<dim3(INPUT / 16, DICT / 16), dim3(16, 16), 0, stream>>>(D, Dt);
  power_kernel<<<1, 1024, 0, stream>>>(D, scal);

  const dim3 residGrid(INPUT / 64, BATCH / 64);   // 8 x 4
  const dim3 updateGrid(DICT / 64, BATCH / 64);   // 16 x 4

  for (int it = 0; it < N_ITERS; ++it) {
    resid_kernel<<<residGrid, 256, 0, stream>>>(x, Dt, h, resid, flags);
    update_kernel<<<updateGrid, 256, 0, stream>>>(resid, D, h, scal, rowsq, flags);
    check_kernel<<<1, 256, 0, stream>>>(rowsq, flags);
  }
}